// GCN_56719338111366
// MI455X (gfx1250) — compile-verified
//
#include <hip/hip_runtime.h>
#include <stdint.h>

typedef __attribute__((ext_vector_type(2))) float v2f;
typedef __attribute__((ext_vector_type(8))) float v8f;

#define NFEAT 128

// ---------------- degree / symmetric norm ----------------
__global__ void k_deg_init(float* __restrict__ deg, int n) {
    int i = blockIdx.x * blockDim.x + threadIdx.x;
    if (i < n) deg[i] = 1.0f;  // self-loop contributes 1 to every node
}

__global__ void k_deg_accum(const long long* __restrict__ dst, float* __restrict__ deg, int e) {
    int i = blockIdx.x * blockDim.x + threadIdx.x;
    if (i < e) atomicAdd(&deg[(size_t)dst[i]], 1.0f);
}

__global__ void k_dis(float* __restrict__ degdis, int n) {
    int i = blockIdx.x * blockDim.x + threadIdx.x;
    if (i < n) degdis[i] = rsqrtf(fmaxf(degdis[i], 1.0f));
}

// ---------------- fp32 WMMA GEMM: C[16-row strip] = A_strip(16x128) @ W(128xNC) ----------------
// One wave per 16x16 output tile; block covers one 16-row strip, waves span columns.
// A fragment layout (V_WMMA_F32_16X16X4_F32): lanes 0-15 -> M=0..15 K={k,k+1};
// lanes 16-31 -> M=0..15 K={k+2,k+3}. B mirrored: lane half selects K pair, lm = N.
template <int NC, bool BIAS>
__global__ __launch_bounds__(256) void k_gemm(const float* __restrict__ A,
                                              const float* __restrict__ W,
                                              const float* __restrict__ bias,
                                              float* __restrict__ C, int nrows) {
    __shared__ float As[16][NFEAT + 4];
    const int tid = threadIdx.x;
    const size_t m0 = (size_t)blockIdx.x * 16;

    // stage 16x128 A tile into LDS (float4 per step)
    for (int i = tid; i < 16 * (NFEAT / 4); i += blockDim.x) {
        int r  = i >> 5;          // 32 float4 per row
        int c4 = (i & 31) << 2;
        if (m0 + r < (size_t)nrows) {
            const float4 v = *(const float4*)(A + (m0 + r) * NFEAT + c4);
            As[r][c4 + 0] = v.x; As[r][c4 + 1] = v.y;
            As[r][c4 + 2] = v.z; As[r][c4 + 3] = v.w;
        }
    }
    __syncthreads();

    const int lane = tid & 31;
    const int wv   = tid >> 5;
    const int n0   = wv * 16;
    const int half = lane >> 4;
    const int lm   = lane & 15;
    const int ncol = n0 + lm;

    v8f acc = {};
#pragma unroll
    for (int k = 0; k < NFEAT; k += 4) {
        const int kr = k + 2 * half;
        v2f a; a.x = As[lm][kr]; a.y = As[lm][kr + 1];
        v2f b;
        b.x = (NC == 128 || ncol < NC) ? W[kr * NC + ncol]       : 0.0f;
        b.y = (NC == 128 || ncol < NC) ? W[(kr + 1) * NC + ncol] : 0.0f;
        acc = __builtin_amdgcn_wmma_f32_16x16x4_f32(
            /*neg_a=*/false, a, /*neg_b=*/false, b,
            /*c_mod=*/(short)0, acc, /*reuse_a=*/false, /*reuse_b=*/false);
    }

    if (NC == 128 || ncol < NC) {
        const float bv = BIAS ? bias[ncol] : 0.0f;
#pragma unroll
        for (int r = 0; r < 8; ++r) {
            const size_t row = m0 + r + 8 * half;  // C/D layout: VGPR r -> M=r (lo lanes), M=r+8 (hi)
            if (row < (size_t)nrows) C[row * NC + ncol] = acc[r] + bv;
        }
    }
}

// ---------------- aggregation ----------------
// agg[i] = h[i] * dis[i]^2   (self-loop message), float4 per thread
__global__ void k_selfloop(const float* __restrict__ h, const float* __restrict__ dis,
                           float* __restrict__ agg, int n) {
    int i = blockIdx.x * blockDim.x + threadIdx.x;
    if (i >= n * (NFEAT / 4)) return;
    const int node = i >> 5;
    float w = dis[node]; w *= w;
    float4 v = ((const float4*)h)[i];
    float4 o = make_float4(v.x * w, v.y * w, v.z * w, v.w * w);
    ((float4*)agg)[i] = o;
}

// one wave per edge: 32 lanes x float4 = full 128-float row; gathers are L2-resident (51MB < 192MB)
__global__ __launch_bounds__(256) void k_scatter(const float* __restrict__ h,
                                                 const long long* __restrict__ src,
                                                 const long long* __restrict__ dst,
                                                 const float* __restrict__ dis,
                                                 float* __restrict__ agg, int e) {
    const int edge = (int)((blockIdx.x * (size_t)blockDim.x + threadIdx.x) >> 5);
    const int lane = threadIdx.x & 31;
    if (edge >= e) return;
    const long long s = src[edge];
    const long long d = dst[edge];
    const float w = dis[s] * dis[d];
    const float4 v = *(const float4*)(h + (size_t)s * NFEAT + lane * 4);
    float* a = agg + (size_t)d * NFEAT + lane * 4;
    atomicAdd(a + 0, v.x * w);
    atomicAdd(a + 1, v.y * w);
    atomicAdd(a + 2, v.z * w);
    atomicAdd(a + 3, v.w * w);
}

__global__ void k_bias_relu(float* __restrict__ agg, const float* __restrict__ bias, int n) {
    int i = blockIdx.x * blockDim.x + threadIdx.x;
    if (i >= n * (NFEAT / 4)) return;
    const int c4 = (i & 31) << 2;
    float4 v = ((float4*)agg)[i];
    v.x = fmaxf(v.x + bias[c4 + 0], 0.0f);
    v.y = fmaxf(v.y + bias[c4 + 1], 0.0f);
    v.z = fmaxf(v.z + bias[c4 + 2], 0.0f);
    v.w = fmaxf(v.w + bias[c4 + 3], 0.0f);
    ((float4*)agg)[i] = v;
}

// ---------------- launch ----------------
extern "C" void kernel_launch(void* const* d_in, const int* in_sizes, int n_in,
                              void* d_out, int out_size, void* d_ws, size_t ws_size,
                              hipStream_t stream) {
    const float*     x   = (const float*)d_in[0];
    const long long* ei  = (const long long*)d_in[1];
    const float*     W1  = (const float*)d_in[2];
    const float*     b1  = (const float*)d_in[3];
    const float*     W2  = (const float*)d_in[4];
    const float*     b2  = (const float*)d_in[5];
    const float*     Wl  = (const float*)d_in[6];
    const float*     bl  = (const float*)d_in[7];
    float*           out = (float*)d_out;

    const int n = in_sizes[0] / NFEAT;   // 100000
    const int e = in_sizes[1] / 2;       // 1600000
    const long long* src = ei;
    const long long* dst = ei + e;

    char* ws = (char*)d_ws;
    float* dis = (float*)ws;
    size_t off = (((size_t)n * sizeof(float)) + 255) & ~(size_t)255;
    float* bufH = (float*)(ws + off);                 // n*128 f32 (51.2 MB)
    float* bufA = bufH + (size_t)n * NFEAT;           // n*128 f32 (51.2 MB)

    const int nt4     = n * (NFEAT / 4);
    const int mblocks = (n + 15) / 16;                // 6250
    const int eblocks = (e + 7) / 8;                  // 8 edges (waves) per 256-thr block

    // symmetric normalization coefficients
    k_deg_init <<<(n + 255) / 256, 256, 0, stream>>>(dis, n);
    k_deg_accum<<<(e + 255) / 256, 256, 0, stream>>>(dst, dis, e);
    k_dis      <<<(n + 255) / 256, 256, 0, stream>>>(dis, n);

    // ---- layer 1: h = relu(Ahat (x@W1) + b1) ----
    k_gemm<128, false><<<mblocks, 256, 0, stream>>>(x, W1, nullptr, bufH, n);
    k_selfloop <<<(nt4 + 255) / 256, 256, 0, stream>>>(bufH, dis, bufA, n);
    k_scatter  <<<eblocks, 256, 0, stream>>>(bufH, src, dst, dis, bufA, e);
    k_bias_relu<<<(nt4 + 255) / 256, 256, 0, stream>>>(bufA, b1, n);

    // ---- layer 2: h = relu(Ahat (h@W2) + b2) ----
    k_gemm<128, false><<<mblocks, 256, 0, stream>>>(bufA, W2, nullptr, bufH, n);
    k_selfloop <<<(nt4 + 255) / 256, 256, 0, stream>>>(bufH, dis, bufA, n);
    k_scatter  <<<eblocks, 256, 0, stream>>>(bufH, src, dst, dis, bufA, e);
    k_bias_relu<<<(nt4 + 255) / 256, 256, 0, stream>>>(bufA, b2, n);

    // ---- classifier: out = h @ Wl + bl  (NC=40 -> 3 column tiles, last masked) ----
    k_gemm<40, true><<<mblocks, 96, 0, stream>>>(bufA, Wl, bl, out, n);
}